// GNN_32203664785441
// MI455X (gfx1250) — compile-verified
//
#include <hip/hip_runtime.h>

typedef float v2f __attribute__((ext_vector_type(2)));
typedef float v8f __attribute__((ext_vector_type(8)));

#define N_NODES 100000
#define N_EDGES 3200000
#define F_IN    640
#define F_HID   16

// ---------------- utility kernels ----------------

__global__ void fill_zero_f32v4(float4* __restrict__ p, int n4) {
  int i = blockIdx.x * blockDim.x + threadIdx.x;
  if (i < n4) p[i] = make_float4(0.f, 0.f, 0.f, 0.f);
}

__global__ void count_deg(const int* __restrict__ edges, int* __restrict__ deg) {
  int e = blockIdx.x * blockDim.x + threadIdx.x;
  if (e < N_EDGES) atomicAdd(&deg[edges[2 * e + 1]], 1);
}

__global__ void compute_dis(const int* __restrict__ deg, float* __restrict__ dis) {
  int i = blockIdx.x * blockDim.x + threadIdx.x;
  if (i < N_NODES) dis[i] = rsqrtf((float)(deg[i] + 1));  // +1 self loop, always > 0
}

// ---------------- GEMM1: h1[N,16] = x[N,640] @ W1[640,16] ----------------
// one wave -> 16 output rows; block = 8 waves = 128 rows; W1 staged in LDS.
__global__ void __launch_bounds__(256) gemm_x_w1(const float* __restrict__ x,
                                                 const float* __restrict__ W1,
                                                 float* __restrict__ h1) {
  __shared__ float w[F_IN * F_HID];  // 40 KB
  int tid = threadIdx.x;
  for (int i = tid; i < F_IN * F_HID; i += 256) w[i] = W1[i];
  __syncthreads();

  int lane = tid & 31, wave = tid >> 5;
  int rowBase = blockIdx.x * 128 + wave * 16;
  int mrow = lane & 15;           // M (or N) index within tile
  int koff = (lane >> 4) * 2;     // lanes 16-31 hold K+2/K+3 (ISA 7.12.2)
  int r = rowBase + mrow; if (r >= N_NODES) r = N_NODES - 1;  // clamp reads, no divergence
  const float* xrow = x + (size_t)r * F_IN + koff;

  v8f c = {};
  for (int k0 = 0; k0 < F_IN; k0 += 32) {
    __builtin_prefetch(xrow + k0 + 256, 0, 0);       // global_prefetch_b8, ~1KB ahead
    #pragma unroll
    for (int t = 0; t < 8; ++t) {
      int k = k0 + 4 * t;
      v2f a = *(const v2f*)(xrow + k);               // A: VGPR0=K+koff, VGPR1=K+koff+1
      v2f b;
      b.x = w[(k + koff) * F_HID + mrow];            // B: row K, col = lane&15
      b.y = w[(k + koff + 1) * F_HID + mrow];
      c = __builtin_amdgcn_wmma_f32_16x16x4_f32(false, a, false, b, (short)0, c, false, false);
    }
  }

  int rv = rowBase + (lane >> 4) * 8;                // C: VGPR v -> M=v (lanes<16) / v+8
  if (rowBase + 16 <= N_NODES) {                     // wave-uniform fast path
    #pragma unroll
    for (int v = 0; v < 8; ++v)
      h1[(size_t)(rv + v) * F_HID + mrow] = c[v];
  } else {
    #pragma unroll
    for (int v = 0; v < 8; ++v) {
      int rr = rv + v;
      if (rr < N_NODES) h1[(size_t)rr * F_HID + mrow] = c[v];
    }
  }
}

// ---------------- edge aggregation in 16-dim feature space ----------------
// 16 lanes per edge, lane = feature; gather h[row], atomic scatter to agg[col].
__global__ void __launch_bounds__(256) aggregate16(const float* __restrict__ h,
                                                   const int2* __restrict__ edges,
                                                   const float* __restrict__ dis,
                                                   float* __restrict__ agg) {
  int t = blockIdx.x * blockDim.x + threadIdx.x;
  int e = t >> 4;
  int f = t & 15;
  if (e >= N_EDGES) return;
  int2 rc = edges[e];                                // single b64, broadcast across 16 lanes
  float nrm = dis[rc.x] * dis[rc.y];
  float val = nrm * h[(size_t)rc.x * F_HID + f];
  atomicAdd(&agg[(size_t)rc.y * F_HID + f], val);
}

// out = (agg + dis^2 * h_selfloop [+ bias]), optional relu
__global__ void finalize16(const float* __restrict__ agg, const float* __restrict__ h,
                           const float* __restrict__ dis, const float* __restrict__ bias,
                           float* __restrict__ out, int do_relu, int use_bias) {
  int i = blockIdx.x * blockDim.x + threadIdx.x;
  if (i >= N_NODES * F_HID) return;
  int n = i >> 4, f = i & 15;
  float d = dis[n];
  float v = agg[i] + d * d * h[i];
  if (use_bias) v += bias[f];
  out[i] = do_relu ? fmaxf(v, 0.0f) : v;
}

// ---------------- GEMM2: out[N,16] = A[N,16] @ W2[16,16] ----------------
__global__ void __launch_bounds__(256) gemm16(const float* __restrict__ A,
                                              const float* __restrict__ W2,
                                              float* __restrict__ out) {
  __shared__ float w[F_HID * F_HID];
  int tid = threadIdx.x;
  if (tid < F_HID * F_HID) w[tid] = W2[tid];
  __syncthreads();

  int lane = tid & 31, wave = tid >> 5;
  int rowBase = blockIdx.x * 128 + wave * 16;
  int mrow = lane & 15;
  int koff = (lane >> 4) * 2;
  int r = rowBase + mrow; if (r >= N_NODES) r = N_NODES - 1;
  const float* arow = A + (size_t)r * F_HID + koff;

  v8f c = {};
  #pragma unroll
  for (int k = 0; k < F_HID; k += 4) {
    v2f a = *(const v2f*)(arow + k);
    v2f b;
    b.x = w[(k + koff) * F_HID + mrow];
    b.y = w[(k + koff + 1) * F_HID + mrow];
    c = __builtin_amdgcn_wmma_f32_16x16x4_f32(false, a, false, b, (short)0, c, false, false);
  }

  int rv = rowBase + (lane >> 4) * 8;
  if (rowBase + 16 <= N_NODES) {
    #pragma unroll
    for (int v = 0; v < 8; ++v)
      out[(size_t)(rv + v) * F_HID + mrow] = c[v];
  } else {
    #pragma unroll
    for (int v = 0; v < 8; ++v) {
      int rr = rv + v;
      if (rr < N_NODES) out[(size_t)rr * F_HID + mrow] = c[v];
    }
  }
}

// ---------------- GEMM3: out[N,640] = A[N,16] @ W3[16,640] + b3 ----------------
// A tile held in 8 VGPRs, reused across 40 column tiles; W3/b3 staged in LDS.
__global__ void __launch_bounds__(256) gemm_w3(const float* __restrict__ A,
                                               const float* __restrict__ W3,
                                               const float* __restrict__ b3,
                                               float* __restrict__ out) {
  __shared__ float w[F_HID * F_IN];  // 40 KB
  __shared__ float bb[F_IN];
  int tid = threadIdx.x;
  for (int i = tid; i < F_HID * F_IN; i += 256) w[i] = W3[i];
  for (int i = tid; i < F_IN; i += 256) bb[i] = b3[i];
  __syncthreads();

  int lane = tid & 31, wave = tid >> 5;
  int rowBase = blockIdx.x * 128 + wave * 16;
  int mrow = lane & 15;
  int koff = (lane >> 4) * 2;
  int r = rowBase + mrow; if (r >= N_NODES) r = N_NODES - 1;

  v2f a[4];
  #pragma unroll
  for (int t = 0; t < 4; ++t)
    a[t] = *(const v2f*)(A + (size_t)r * F_HID + 4 * t + koff);

  int rv = rowBase + (lane >> 4) * 8;
  bool full = (rowBase + 16 <= N_NODES);             // wave-uniform

  for (int j = 0; j < F_IN / 16; ++j) {
    int n0 = j * 16;
    v8f c = {};
    #pragma unroll
    for (int t = 0; t < 4; ++t) {
      int kb = 4 * t + koff;
      v2f b;
      b.x = w[kb * F_IN + n0 + mrow];
      b.y = w[(kb + 1) * F_IN + n0 + mrow];
      c = __builtin_amdgcn_wmma_f32_16x16x4_f32(false, a[t], false, b, (short)0, c, false, false);
    }
    float bias = bb[n0 + mrow];
    if (full) {
      #pragma unroll
      for (int v = 0; v < 8; ++v)
        out[(size_t)(rv + v) * F_IN + n0 + mrow] = c[v] + bias;
    } else {
      #pragma unroll
      for (int v = 0; v < 8; ++v) {
        int rr = rv + v;
        if (rr < N_NODES) out[(size_t)rr * F_IN + n0 + mrow] = c[v] + bias;
      }
    }
  }
}

// ---------------- launch ----------------

extern "C" void kernel_launch(void* const* d_in, const int* in_sizes, int n_in,
                              void* d_out, int out_size, void* d_ws, size_t ws_size,
                              hipStream_t stream) {
  const float* x   = (const float*)d_in[0];
  const int* edges = (const int*)d_in[1];
  const float* W1  = (const float*)d_in[2];
  const float* b1  = (const float*)d_in[3];
  const float* W2  = (const float*)d_in[4];
  const float* b2  = (const float*)d_in[5];
  const float* W3  = (const float*)d_in[6];
  const float* b3  = (const float*)d_in[7];
  float* out = (float*)d_out;

  // workspace layout (floats): deg | dis | bufA | bufB | bufC  (~20 MB total)
  int*   deg  = (int*)d_ws;
  float* dis  = (float*)d_ws + N_NODES;
  float* bufA = (float*)d_ws + 2 * N_NODES;
  float* bufB = bufA + (size_t)N_NODES * F_HID;
  float* bufC = bufB + (size_t)N_NODES * F_HID;

  const int nh = N_NODES * F_HID;
  const int gemmBlocks = (N_NODES + 127) / 128;
  const int aggBlocks  = (N_EDGES * 16 + 255) / 256;
  const int zeroBlocksDeg = (N_NODES / 4 + 255) / 256;
  const int zeroBlocksH   = (nh / 4 + 255) / 256;

  // degrees -> dis (shared by all layers)
  fill_zero_f32v4<<<zeroBlocksDeg, 256, 0, stream>>>((float4*)deg, N_NODES / 4);
  count_deg<<<(N_EDGES + 255) / 256, 256, 0, stream>>>(edges, deg);
  compute_dis<<<(N_NODES + 255) / 256, 256, 0, stream>>>(deg, dis);

  // layer 1: h1 = x @ W1 ; aggregate ; relu
  gemm_x_w1<<<gemmBlocks, 256, 0, stream>>>(x, W1, bufA);
  fill_zero_f32v4<<<zeroBlocksH, 256, 0, stream>>>((float4*)bufB, nh / 4);
  aggregate16<<<aggBlocks, 256, 0, stream>>>(bufA, (const int2*)edges, dis, bufB);
  finalize16<<<(nh + 255) / 256, 256, 0, stream>>>(bufB, bufA, dis, b1, bufC, 1, 1);

  // layer 2: h2 = h @ W2 ; aggregate ; relu
  gemm16<<<gemmBlocks, 256, 0, stream>>>(bufC, W2, bufA);
  fill_zero_f32v4<<<zeroBlocksH, 256, 0, stream>>>((float4*)bufB, nh / 4);
  aggregate16<<<aggBlocks, 256, 0, stream>>>(bufA, (const int2*)edges, dis, bufB);
  finalize16<<<(nh + 255) / 256, 256, 0, stream>>>(bufB, bufA, dis, b2, bufC, 1, 1);

  // layer 3: aggregate FIRST in 16-dim (A_hat @ H) @ W3 == A_hat @ (H @ W3)
  fill_zero_f32v4<<<zeroBlocksH, 256, 0, stream>>>((float4*)bufB, nh / 4);
  aggregate16<<<aggBlocks, 256, 0, stream>>>(bufC, (const int2*)edges, dis, bufB);
  finalize16<<<(nh + 255) / 256, 256, 0, stream>>>(bufB, bufC, dis, b1, bufA, 0, 0);
  gemm_w3<<<gemmBlocks, 256, 0, stream>>>(bufA, W3, b3, out);
}